// bigcnn_lut_77180562309381
// MI455X (gfx1250) — compile-verified
//
#include <hip/hip_runtime.h>
#include <math.h>

typedef __attribute__((ext_vector_type(2))) float v2f;
typedef __attribute__((ext_vector_type(8))) float v8f;

#define NUM_CLASSES 100
#define BATCH 512
#define ROWS (BATCH * 8 * 10)   // 40960 rows of the big GEMM
#define NCOL 4096
#define NT   (NCOL / 16)        // 256 N-tiles

// sign of the first-stage projection at (row, k)   [k < 45 assumed]
__device__ __forceinline__ float sign_at(const float* __restrict__ x,
                                         const float* __restrict__ S1,
                                         const float* __restrict__ T1,
                                         int b, int p, int c2, int k) {
  int flat = c2 * 45 + k;          // position within the 450-block of (b,p)
  int c = flat / 15;
  int kk = flat - c * 15;
  const float* xp = x + b * 480 + p * 60 + c * 2;
  float hv = xp[0] * S1[(2 * c) * 15 + kk]
           + xp[1] * S1[(2 * c + 1) * 15 + kk]
           - T1[c * 15 + kk] - 1e-4f;
  return hv > 0.0f ? 1.0f : (hv < 0.0f ? -1.0f : 0.0f);
}

// Kernel 1: fused sign-projection + (40960x45)x(45x4096) fp32 WMMA GEMM + row argmax.
// B-tiles (48x16 of H1, K zero-padded) are staged once per block into LDS with
// async global->LDS copies (double buffered), shared by all 8 waves.
__global__ __launch_bounds__(256)
void gemm_argmax_kernel(const float* __restrict__ x,
                        const float* __restrict__ S1,
                        const float* __restrict__ T1,
                        const float* __restrict__ H1,
                        int* __restrict__ idx_out) {
  __shared__ float smem[2][48 * 16];   // 2 x 3KB B-tile buffers (K padded to 48)

  const int t    = threadIdx.x;
  const int lane = t & 31;
  const int wave = t >> 5;
  const int mt   = blockIdx.x * 8 + wave;      // 0..2559 M-tiles (one per wave)
  const int baseRow = mt * 16;
  const int m   = lane & 15;
  const int hi  = lane >> 4;                   // 0: K+0/1 & rows 0..7 ; 1: K+2/3 & rows 8..15
  const int nl  = lane & 15;                   // N within tile
  const int row = baseRow + m;
  const int b   = row / 80;
  const int p   = (row / 10) & 7;
  const int c2  = row % 10;

  // ---- zero the K = 45..47 pad rows of both LDS buffers (once) ----
  if (t < 96) {
    int bufi = t / 48;
    int r    = t % 48;
    smem[bufi][45 * 16 + r] = 0.0f;
  }

  // ---- async load of one B-tile (rows 0..44, 16 cols) into smem[bufi] ----
  // 45 rows x 64B = 180 b128 lane-transfers, spread over threads 0..179.
  auto issue_tile = [&](int nt, int bufi) {
    if (t < 180) {
      int k   = t >> 2;          // 0..44
      int off = t & 3;           // 4 floats (16B) each
      uint64_t ga = (uint64_t)(uintptr_t)(H1 + (size_t)k * NCOL + nt * 16 + off * 4);
      uint32_t la = (uint32_t)(uintptr_t)(&smem[bufi][k * 16 + off * 4]);
      asm volatile("global_load_async_to_lds_b128 %0, %1, off"
                   :: "v"(la), "v"(ga) : "memory");
    }
  };

  // ---- Build A tile (16 x 48, K zero-padded) in WMMA A layout: 12 x v2f ----
  v2f a[12];
#pragma unroll
  for (int q = 0; q < 11; ++q) {
    int k0 = 4 * q + 2 * hi;                   // <= 42, k0+1 <= 43 : always valid
    a[q].x = sign_at(x, S1, T1, b, p, c2, k0);
    a[q].y = sign_at(x, S1, T1, b, p, c2, k0 + 1);
  }
  {
    float s44 = sign_at(x, S1, T1, b, p, c2, 44);
    a[11].x = hi ? 0.0f : s44;                 // K=44 (hi=0) / K=46 (pad)
    a[11].y = 0.0f;                            // K=45 / K=47 (pad)
  }

  float best[8];
  int   bidx[8];
#pragma unroll
  for (int r = 0; r < 8; ++r) { best[r] = -1.0e30f; bidx[r] = 0; }

  // ---- prologue: stage tile 0 ----
  issue_tile(0, 0);
  asm volatile("s_wait_asynccnt 0x0" ::: "memory");
  __syncthreads();

  // ---- sweep all 256 N-tiles, double-buffered ----
  for (int nt = 0; nt < NT; ++nt) {
    const int cur = nt & 1;
    if (nt + 1 < NT) issue_tile(nt + 1, 1 - cur);   // overlap with compute below

    const float* __restrict__ sbuf = &smem[cur][0];
    const int n = nt * 16 + nl;

    v8f acc = {};
#pragma unroll
    for (int q = 0; q < 12; ++q) {
      int k0 = 4 * q + 2 * hi;
      v2f bq;
      bq.x = sbuf[k0 * 16 + nl];          // pad rows are zero -> no selects needed
      bq.y = sbuf[(k0 + 1) * 16 + nl];
      acc = __builtin_amdgcn_wmma_f32_16x16x4_f32(
          /*neg_a=*/false, a[q], /*neg_b=*/false, bq,
          /*c_mod=*/(short)0, acc, /*reuse_a=*/false, /*reuse_b=*/false);
    }

    // running per-lane argmax: lane holds column n for rows r + 8*hi
#pragma unroll
    for (int r = 0; r < 8; ++r) {
      float v = acc[r];
      if (v > best[r]) { best[r] = v; bidx[r] = n; }   // strict > keeps first max
    }

    asm volatile("s_wait_asynccnt 0x0" ::: "memory");  // own async writes done
    __syncthreads();                                   // tile visible / buffer reusable
  }

  // ---- reduce across the 16 lanes that share each row; first-occurrence ties ----
#pragma unroll
  for (int r = 0; r < 8; ++r) {
    float v = best[r];
    int   i = bidx[r];
#pragma unroll
    for (int off = 1; off < 16; off <<= 1) {
      float v2 = __shfl_xor(v, off, 32);
      int   i2 = __shfl_xor(i, off, 32);
      if (v2 > v || (v2 == v && i2 < i)) { v = v2; i = i2; }
    }
    if ((lane & 15) == 0)
      idx_out[baseRow + r + 8 * hi] = i;
  }
}

// Kernel 2: LUT1 gather + second stage + LUT2 gather + log_softmax. 1 wave / batch row.
__global__ __launch_bounds__(32)
void finish_kernel(const int* __restrict__ idx1,
                   const float* __restrict__ LUT1,
                   const float* __restrict__ S2,
                   const float* __restrict__ H2,
                   const float* __restrict__ T2,
                   const float* __restrict__ LUT2,
                   float* __restrict__ out) {
  const int bb = blockIdx.x;          // batch row
  const int l  = threadIdx.x;         // 0..31

  // z[b, p, e]  held by lane t = p*2+e (t<16); lanes 16..31 compute duplicates
  float z = 0.0f;
  {
    int pp = (l >> 1) & 7;
    int e  = l & 1;
    const int* rowi = idx1 + (bb * 8 + pp) * 10;
#pragma unroll
    for (int c2 = 0; c2 < 10; ++c2)
      z += LUT1[((size_t)c2 * 4096 + rowi[c2]) * 2 + e];
  }

  // per-c (=p) second stage; lane c (and its duplicates) computes idx2[c]
  const int c = l & 7;
  float za = __shfl(z, 2 * c, 32);
  float zb = __shfl(z, 2 * c + 1, 32);

  float s[15];
#pragma unroll
  for (int k = 0; k < 15; ++k) {
    float w = za * S2[(c * 2) * 15 + k]
            + zb * S2[(c * 2 + 1) * 15 + k]
            - T2[c * 15 + k];
    s[k] = (w > 0.0f) ? 1.0f : -1.0f;    // where(w==0,-1,sign(w))
  }

  float bmax = -1.0e30f;
  int   bi   = 0;
#pragma unroll
  for (int k2 = 0; k2 < 16; ++k2) {
    float acc = 0.0f;
#pragma unroll
    for (int d = 0; d < 15; ++d) acc += s[d] * H2[d * 16 + k2];
    if (acc > bmax) { bmax = acc; bi = k2; }   // first-occurrence argmax
  }

  // broadcast idx2[0..7]
  int i2s[8];
#pragma unroll
  for (int c0 = 0; c0 < 8; ++c0) i2s[c0] = __shfl(bi, c0, 32);

  // logits: lane l covers classes l, l+32, l+64, l+96
  float lg[4];
#pragma unroll
  for (int j = 0; j < 4; ++j) {
    int cls = l + 32 * j;
    float acc = 0.0f;
    if (cls < NUM_CLASSES) {
#pragma unroll
      for (int c0 = 0; c0 < 8; ++c0)
        acc += LUT2[((size_t)c0 * 16 + i2s[c0]) * NUM_CLASSES + cls];
    }
    lg[j] = acc;
  }

  // log_softmax over 100 classes
  float mx = -1.0e30f;
#pragma unroll
  for (int j = 0; j < 4; ++j) {
    int cls = l + 32 * j;
    if (cls < NUM_CLASSES && lg[j] > mx) mx = lg[j];
  }
#pragma unroll
  for (int off = 1; off < 32; off <<= 1) {
    float m2 = __shfl_xor(mx, off, 32);
    if (m2 > mx) mx = m2;
  }
  float se = 0.0f;
#pragma unroll
  for (int j = 0; j < 4; ++j) {
    int cls = l + 32 * j;
    if (cls < NUM_CLASSES) se += __expf(lg[j] - mx);
  }
#pragma unroll
  for (int off = 1; off < 32; off <<= 1)
    se += __shfl_xor(se, off, 32);

  float lse = logf(se) + mx;
#pragma unroll
  for (int j = 0; j < 4; ++j) {
    int cls = l + 32 * j;
    if (cls < NUM_CLASSES)
      out[(size_t)bb * NUM_CLASSES + cls] = lg[j] - lse;
  }
}

extern "C" void kernel_launch(void* const* d_in, const int* in_sizes, int n_in,
                              void* d_out, int out_size, void* d_ws, size_t ws_size,
                              hipStream_t stream) {
  const float* x    = (const float*)d_in[0];
  const float* S1   = (const float*)d_in[1];
  const float* H1   = (const float*)d_in[2];
  const float* T1   = (const float*)d_in[3];
  const float* LUT1 = (const float*)d_in[4];
  const float* S2   = (const float*)d_in[5];
  const float* H2   = (const float*)d_in[6];
  const float* T2   = (const float*)d_in[7];
  const float* LUT2 = (const float*)d_in[8];
  float* out = (float*)d_out;

  int* idx1 = (int*)d_ws;   // ROWS ints = 160 KB

  // 2560 M-tiles, 8 waves (1 tile each) per 256-thread block
  gemm_argmax_kernel<<<dim3(ROWS / 16 / 8), dim3(256), 0, stream>>>(x, S1, T1, H1, idx1);
  finish_kernel<<<dim3(BATCH), dim3(32), 0, stream>>>(idx1, LUT1, S2, H2, T2, LUT2, out);
}